// ATIN_op_10926396801590
// MI455X (gfx1250) — compile-verified
//
#include <hip/hip_runtime.h>
#include <hip/hip_bf16.h>

typedef __attribute__((ext_vector_type(2))) float v2f;
typedef __attribute__((ext_vector_type(8))) float v8f;

#define B_    32
#define V_    64
#define L_    2048
#define C_    64
#define K_    3
#define H_    32
#define LP_   2046   // L - K + 1
#define TOPK_ 1023   // LP_ * 0.5
#define NPAD_ 2048   // sort width (power of two)

// ---------------------------------------------------------------------------
// Hardware tanh (CDNA5 V_TANH_F32 trans op) with toolchain fallbacks.
// ---------------------------------------------------------------------------
__device__ __forceinline__ float hw_tanh(float x) {
#if defined(__has_builtin) && __has_builtin(__builtin_amdgcn_tanhf)
    return __builtin_amdgcn_tanhf(x);
#elif defined(__has_builtin) && __has_builtin(__builtin_amdgcn_tanh_f32)
    return __builtin_amdgcn_tanh_f32(x);
#else
    float r;
    // TRANS op: cover the result-latency hazard explicitly (ISA 7.4.3).
    asm volatile("v_tanh_f32 %0, %1\n\tv_nop\n\tv_nop" : "=v"(r) : "v"(x));
    return r;
#endif
}

// sigmoid(z) = 0.5*(1 + tanh(z/2)) -- single trans op, monotone in z
__device__ __forceinline__ float hw_sigmoid(float z) {
    return 0.5f * (1.0f + hw_tanh(0.5f * z));
}

// ---------------------------------------------------------------------------
// Kernel 1: fold conv(1->64,k=3) + Linear(64->32) into per-variable [32 x 4]
// filter (k=3 zero-padded to 4 for WMMA K=4) and fused bias.
// A4[v][j][k] = sum_c W1[j,c] * conv_w[v,c,k];  B1v[v][j] = W1.conv_b + b1[j]
// ---------------------------------------------------------------------------
__global__ __launch_bounds__(256) void fold_weights_kernel(
    const float* __restrict__ conv_w,  // [V,C,K]
    const float* __restrict__ conv_b,  // [V,C]
    const float* __restrict__ W1,      // [H,C]
    const float* __restrict__ b1,      // [H]
    float* __restrict__ A4,            // [V,H,4]
    float* __restrict__ B1v)           // [V,H]
{
    int idx = blockIdx.x * blockDim.x + threadIdx.x;  // v*H + j
    if (idx >= V_ * H_) return;
    int v = idx >> 5, j = idx & (H_ - 1);
    float s0 = 0.f, s1 = 0.f, s2 = 0.f, sb = 0.f;
    const float* w1r = W1 + j * C_;
    const float* cwv = conv_w + v * C_ * K_;
    const float* cbv = conv_b + v * C_;
    #pragma unroll 4
    for (int c = 0; c < C_; ++c) {
        float w = w1r[c];
        s0 += w * cwv[c * K_ + 0];
        s1 += w * cwv[c * K_ + 1];
        s2 += w * cwv[c * K_ + 2];
        sb += w * cbv[c];
    }
    float* a = A4 + idx * 4;
    a[0] = s0; a[1] = s1; a[2] = s2; a[3] = 0.f;   // zero-pad K=3 -> 4
    B1v[idx] = sb + b1[j];
}

// ---------------------------------------------------------------------------
// Kernel 2: scores via V_WMMA_F32_16X16X4_F32.
// Per wave: one (row, 16-position) tile.  H[32x16] = A4_v[32x4] x Xwin[4x16],
// two WMMAs (M rows 0-15 and 16-31).  Then per-lane hw-tanh/W2 reduction,
// cross-half shuffle, hw-sigmoid, store score.
//
// Operand layouts (ISA 7.12.2, f32, wave32):
//   A 16x4 : lane L (M = L%16): VGPR0 = K=base, VGPR1 = K=base+1, base = (L<16?0:2)
//   B 4x16 : lane L (N = L%16): VGPR r holds K-row (r + (L<16?0:2))
//   C/D    : lane L (N = L%16): VGPR r holds M = r + (L<16?0:8)
// ---------------------------------------------------------------------------
__global__ __launch_bounds__(256) void score_kernel(
    const float* __restrict__ x,     // [B,V,L]
    const float* __restrict__ A4,    // [V,H,4]
    const float* __restrict__ B1v,   // [V,H]
    const float* __restrict__ W2,    // [1,H]
    const float* __restrict__ b2,    // [1]
    float* __restrict__ scores)      // [B*V, NPAD_]
{
    const int row  = blockIdx.x;            // b*V + v
    const int v    = row & (V_ - 1);
    const int lane = threadIdx.x & 31;
    const int wave = threadIdx.x >> 5;
    const int tile = blockIdx.y * 8 + wave; // 0..127
    const int t0   = tile * 16;

    const int n      = lane & 15;
    const int hiHalf = lane >> 4;           // 0 or 1
    const int kbase  = hiHalf * 2;

    // A operands for the two M-tiles (rows 0-15 and 16-31)
    const float* Av = A4 + v * H_ * 4;
    v2f a0, a1;
    a0.x = Av[n * 4 + kbase];          a0.y = Av[n * 4 + kbase + 1];
    a1.x = Av[(16 + n) * 4 + kbase];   a1.y = Av[(16 + n) * 4 + kbase + 1];

    // B operand: B[k][n] = x[row, t0 + n + k]  (k = kbase, kbase+1); bounds-guard
    const float* xr = x + row * L_;
    int i0 = t0 + n + kbase;
    int i1 = i0 + 1;
    v2f bm;
    bm.x = (i0 < L_) ? xr[i0] : 0.f;
    bm.y = (i1 < L_) ? xr[i1] : 0.f;

    v8f c0 = {}; v8f c1 = {};
    c0 = __builtin_amdgcn_wmma_f32_16x16x4_f32(false, a0, false, bm,
                                               (short)0, c0, false, false);
    c1 = __builtin_amdgcn_wmma_f32_16x16x4_f32(false, a1, false, bm,
                                               (short)0, c1, false, false);

    // Per-lane partial of z[n] = sum_j W2[j]*tanh(H[j][n] + B1v[v][j])
    const float* Bv = B1v + v * H_;
    float partial = 0.f;
    #pragma unroll
    for (int r = 0; r < 8; ++r) {
        int j0 = r + hiHalf * 8;   // rows of c0 tile this lane holds
        int j1 = 16 + j0;          // rows of c1 tile this lane holds
        partial += W2[j0] * hw_tanh(c0[r] + Bv[j0]);
        partial += W2[j1] * hw_tanh(c1[r] + Bv[j1]);
    }
    // combine lane n and lane n+16 -> full 32-row dot product
    partial += __shfl_xor(partial, 16, 32);

    float sc = hw_sigmoid(partial + b2[0]);

    int t = t0 + n;
    if (hiHalf == 0 && t < LP_) scores[row * NPAD_ + t] = sc;
}

// ---------------------------------------------------------------------------
// Kernel 3: per-row bitonic sort of 2048 packed 64-bit keys in LDS, then emit
// the top-1023 indices in descending-score order (stable: ties -> lower index,
// matching jax.lax.top_k).  key = sortable(score)<<32 | ~t ; padding key = 0.
// ---------------------------------------------------------------------------
__global__ __launch_bounds__(256) void topk_sort_kernel(
    const float* __restrict__ scores,  // [B*V, NPAD_]
    int* __restrict__ out)             // [B*V, TOPK_]
{
    __shared__ unsigned long long keys[NPAD_];
    const int row = blockIdx.x;
    const int tid = threadIdx.x;

    for (int i = tid; i < NPAD_; i += 256) {
        unsigned long long kk = 0ull;
        if (i < LP_) {
            unsigned u = __float_as_uint(scores[row * NPAD_ + i]);
            u = (u & 0x80000000u) ? ~u : (u | 0x80000000u);  // total-order map
            kk = ((unsigned long long)u << 32) | (unsigned)(~i);
        }
        keys[i] = kk;
    }
    __syncthreads();

    // bitonic sort, overall descending
    for (unsigned k = 2; k <= NPAD_; k <<= 1) {
        for (unsigned j = k >> 1; j > 0; j >>= 1) {
            for (unsigned p = tid; p < NPAD_ / 2; p += 256) {
                unsigned i   = 2 * p - (p & (j - 1));
                unsigned ixj = i | j;
                unsigned long long a = keys[i], b = keys[ixj];
                bool desc = ((i & k) == 0);
                bool sw   = desc ? (a < b) : (a > b);
                if (sw) { keys[i] = b; keys[ixj] = a; }
            }
            __syncthreads();
        }
    }

    for (int i = tid; i < TOPK_; i += 256) {
        out[row * TOPK_ + i] = (int)(unsigned)(~keys[i]);  // recover t
    }
}

// ---------------------------------------------------------------------------
extern "C" void kernel_launch(void* const* d_in, const int* in_sizes, int n_in,
                              void* d_out, int out_size, void* d_ws, size_t ws_size,
                              hipStream_t stream) {
    const float* x      = (const float*)d_in[0];
    const float* conv_w = (const float*)d_in[1];
    const float* conv_b = (const float*)d_in[2];
    const float* W1     = (const float*)d_in[3];
    const float* b1     = (const float*)d_in[4];
    const float* W2     = (const float*)d_in[5];
    const float* b2     = (const float*)d_in[6];
    int* out = (int*)d_out;

    char*  ws     = (char*)d_ws;
    float* A4     = (float*)ws;                                         // 32 KB
    float* B1v    = (float*)(ws + (size_t)V_ * H_ * 4 * sizeof(float)); // 8 KB
    float* scores = (float*)(ws + 65536);                               // 16 MB

    fold_weights_kernel<<<dim3((V_ * H_ + 255) / 256), 256, 0, stream>>>(
        conv_w, conv_b, W1, b1, A4, B1v);

    // 2048 rows x (16 tile-groups of 8 waves) covers 128 tiles of 16 positions
    score_kernel<<<dim3(B_ * V_, 16), 256, 0, stream>>>(
        x, A4, B1v, W2, b2, scores);

    topk_sort_kernel<<<dim3(B_ * V_), 256, 0, stream>>>(scores, out);
}